// DeepGCN_Node_81123342287178
// MI455X (gfx1250) — compile-verified
//
#include <hip/hip_runtime.h>

// DeepGCN node: GATv2Conv + edge MLP, fp32, MI455X (gfx1250).
// All GEMMs run on V_WMMA_F32_16X16X4_F32 (full fp32, no downcast).
// Weights are staged in LDS pre-swizzled so every WMMA B-fragment is a single
// conflict-free ds_load_b64 (pair-row stride 160 dwords -> halves hit disjoint
// 32-bank sets; compiler merges pairs into ds_load_2addr_b64). B-fragments are
// batch-loaded per K-step before the WMMA burst so one dscnt wait covers the
// whole load clause. Gathers/atomics are L2-resident (12.8MB << 192MB L2).

constexpr int N = 50000;
constexpr int E = 800000;
constexpr int D = 64;
constexpr int SW = 160;  // LDS pair-row stride (dwords) for swizzled weights
constexpr int SA = 132;  // LDS row stride (dwords) for activation tiles

typedef __attribute__((ext_vector_type(2))) float v2f;
typedef __attribute__((ext_vector_type(8))) float v8f;

__device__ __forceinline__ v8f wmma4(v2f a, v2f b, v8f c) {
  // D = A(16x4) * B(4x16) + C, fp32
  return __builtin_amdgcn_wmma_f32_16x16x4_f32(false, a, false, b, (short)0, c,
                                               false, false);
}

// Stage W (kdim x 64 row-major) into LDS swizzled: a B fragment for K-pair
// p = k>>1, column n is the contiguous 8B pair sWp[p*SW + n*2 .. +1].
__device__ __forceinline__ void stage_w(const float* __restrict__ W,
                                        float* __restrict__ sWp, int kdim,
                                        int tid, int nth) {
  for (int i = tid; i < kdim * 64; i += nth) {
    int k = i >> 6, n = i & 63;
    sWp[(k >> 1) * SW + n * 2 + (k & 1)] = W[i];
  }
}

// Order-preserving float<->int map so float segment-max can use int atomicMax.
__device__ __forceinline__ int ord_of(float f) {
  int i = __float_as_int(f);
  return i < 0 ? (int)0x80000000 - i : i;
}
__device__ __forceinline__ float f_of_ord(int m) {
  int i = m < 0 ? (int)0x80000000 - m : m;
  return __int_as_float(i);
}

// ---------------------------------------------------------------- init
__global__ void k_init(float* __restrict__ xout, int* __restrict__ mx,
                       float* __restrict__ dn, const float* __restrict__ bias) {
  int i = blockIdx.x * 256 + threadIdx.x;
  if (i < N * D) xout[i] = bias[i & 63];
  if (i < N * 4) { mx[i] = (int)0x80000000; dn[i] = 0.f; }
}

// ------------------------------------------------- xl = x@Wl, xr = x@Wr
// One A-fragment load feeds both weight matrices (x read once from HBM).
__global__ void __launch_bounds__(256)
k_node_gemm(const float* __restrict__ x, const float* __restrict__ Wl,
            const float* __restrict__ Wr, float* __restrict__ xl,
            float* __restrict__ xr) {
  __shared__ float sWl[32 * SW];  // 20 KB each
  __shared__ float sWr[32 * SW];
  stage_w(Wl, sWl, 64, threadIdx.x, 256);
  stage_w(Wr, sWr, 64, threadIdx.x, 256);
  __syncthreads();
  const int wave = threadIdx.x >> 5, lane = threadIdx.x & 31;
  const int tile = blockIdx.x * 8 + wave;  // 16 rows per wave-tile
  if (tile * 16 >= N) return;              // wave-uniform exit
  const int row0 = tile * 16, m = lane & 15, half = lane >> 4;
  v8f accl[4] = {}, accr[4] = {};
  for (int kt = 0; kt < 16; ++kt) {
    const int kb = kt * 4 + half * 2;  // per-lane K pair
    v2f a = *(const v2f*)(x + (row0 + m) * D + kb);
    const float* wl = sWl + (kt * 2 + half) * SW + m * 2;
    const float* wr = sWr + (kt * 2 + half) * SW + m * 2;
    v2f bl[4], br[4];
#pragma unroll
    for (int nt = 0; nt < 4; ++nt) {  // batch the DS loads: one clause+wait
      bl[nt] = *(const v2f*)(wl + nt * 32);
      br[nt] = *(const v2f*)(wr + nt * 32);
    }
#pragma unroll
    for (int nt = 0; nt < 4; ++nt) {  // then the WMMA burst
      accl[nt] = wmma4(a, bl[nt], accl[nt]);
      accr[nt] = wmma4(a, br[nt], accr[nt]);
    }
  }
#pragma unroll
  for (int nt = 0; nt < 4; ++nt)
#pragma unroll
    for (int v = 0; v < 8; ++v) {
      const int row = (row0 + v + 8 * half) * D + nt * 16 + m;
      xl[row] = accl[nt][v];
      xr[row] = accr[nt][v];
    }
}

// ---- ef = edge_attr@We (WMMA, kept in regs) -> leaky_relu(xl[s]+xr[d]+ef)
// ---- -> per-head logit (shfl reduce) -> store + segment-max atomic
__global__ void __launch_bounds__(256)
k_edge_logits(const float* __restrict__ ea, const int* __restrict__ ei,
              const float* __restrict__ We, const float* __restrict__ att,
              const float* __restrict__ xl, const float* __restrict__ xr,
              float* __restrict__ lg, int* __restrict__ mx) {
  __shared__ float sW[32 * SW];  // 20 KB
  __shared__ float sA[64];
  stage_w(We, sW, 64, threadIdx.x, 256);
  if (threadIdx.x < 64) sA[threadIdx.x] = att[threadIdx.x];
  __syncthreads();
  const int wave = threadIdx.x >> 5, lane = threadIdx.x & 31;
  const int tile = blockIdx.x * 8 + wave;
  const int e0 = tile * 16;
  if (e0 >= E) return;
  const int m = lane & 15, half = lane >> 4;
  v8f acc[4] = {};
  for (int kt = 0; kt < 16; ++kt) {
    const int kb = kt * 4 + half * 2;
    v2f a = *(const v2f*)(ea + (e0 + m) * D + kb);
    const float* w = sW + (kt * 2 + half) * SW + m * 2;
    v2f b[4];
#pragma unroll
    for (int nt = 0; nt < 4; ++nt) b[nt] = *(const v2f*)(w + nt * 32);
#pragma unroll
    for (int nt = 0; nt < 4; ++nt) acc[nt] = wmma4(a, b[nt], acc[nt]);
  }
  // acc[nt][v] = ef[e0 + v + 8*half][nt*16 + m];  head = nt, channel = m
#pragma unroll
  for (int v = 0; v < 8; ++v) {
    const int e = e0 + v + 8 * half;
    const int s = ei[e], d = ei[E + e];
#pragma unroll
    for (int nt = 0; nt < 4; ++nt) {
      float z = acc[nt][v] + xl[s * D + nt * 16 + m] + xr[d * D + nt * 16 + m];
      z = z >= 0.f ? z : 0.2f * z;
      float p = z * sA[nt * 16 + m];
      p += __shfl_xor(p, 1);
      p += __shfl_xor(p, 2);
      p += __shfl_xor(p, 4);
      p += __shfl_xor(p, 8);  // reduce within each 16-lane half
      if (m == 0) {
        lg[e * 4 + nt] = p;
        atomicMax(&mx[d * 4 + nt], ord_of(p));
      }
    }
  }
}

// -------------------------------------- exp(logit - max), segment-sum denom
__global__ void k_softmax_den(const int* __restrict__ ei, float* __restrict__ lg,
                              const int* __restrict__ mx, float* __restrict__ dn) {
  int i = blockIdx.x * 256 + threadIdx.x;
  if (i >= E * 4) return;
  int e = i >> 2, h = i & 3;
  int d = ei[E + e];
  float v = __expf(lg[i] - f_of_ord(mx[d * 4 + h]));
  lg[i] = v;  // overwrite logits with exp in place
  atomicAdd(&dn[d * 4 + h], v);
}

// ------------------------------- x_out[dst] += alpha * xl[src]  (per e,n)
__global__ void k_scatter(const int* __restrict__ ei, const float* __restrict__ ex,
                          const float* __restrict__ dn, const float* __restrict__ xl,
                          float* __restrict__ xout) {
  int i = blockIdx.x * 256 + threadIdx.x;
  if (i >= E * D) return;
  int e = i >> 6, n = i & 63, h = n >> 4;
  int s = ei[e], d = ei[E + e];
  float alpha = ex[e * 4 + h] / (dn[d * 4 + h] + 1e-16f);
  atomicAdd(&xout[d * D + n], xl[s * D + n] * alpha);
}

// -- gather cat=[x_out[src],x_out[dst]] -> LN -> ReLU -> @Wm+bm -> +edge_attr
__global__ void __launch_bounds__(128)
k_edge_mlp(const int* __restrict__ ei, const float* __restrict__ xout,
           const float* __restrict__ ea, const float* __restrict__ g,
           const float* __restrict__ bvec, const float* __restrict__ Wm,
           const float* __restrict__ bm, float* __restrict__ eout) {
  __shared__ float sW[64 * SW];      // 40 KB (128x64 swizzled)
  __shared__ float sh[4][16 * SA];   // 33 KB (per-wave 16x128 tile, padded)
  __shared__ float sg[128], sb[128], sbm[64];
  stage_w(Wm, sW, 128, threadIdx.x, 128);
  if (threadIdx.x < 128) { sg[threadIdx.x] = g[threadIdx.x]; sb[threadIdx.x] = bvec[threadIdx.x]; }
  if (threadIdx.x < 64) sbm[threadIdx.x] = bm[threadIdx.x];
  __syncthreads();
  const int wave = threadIdx.x >> 5, lane = threadIdx.x & 31;
  const int tile = blockIdx.x * 4 + wave;
  const int e0 = tile * 16;
  if (e0 >= E) return;
  const int r = lane & 15, half = lane >> 4;
  // lane (r, half) owns half-row: half 0 -> src features, half 1 -> dst
  float* hrow = &sh[wave][r * SA + half * 64];
  const int e = e0 + r;
  const int node = half ? ei[E + e] : ei[e];
  const float4* src4 = (const float4*)(xout + node * D);
  float sum = 0.f, sq = 0.f;
#pragma unroll
  for (int k = 0; k < 16; ++k) {
    float4 v = src4[k];
    ((float4*)hrow)[k] = v;
    sum += v.x + v.y + v.z + v.w;
    sq += v.x * v.x + v.y * v.y + v.z * v.z + v.w * v.w;
  }
  sum += __shfl_xor(sum, 16);  // combine the two half-rows
  sq += __shfl_xor(sq, 16);
  const float mu = sum * (1.f / 128.f);
  const float rs = rsqrtf(sq * (1.f / 128.f) - mu * mu + 1e-5f);
#pragma unroll 4
  for (int k = 0; k < 64; ++k) {
    int idx = half * 64 + k;
    float vv = (hrow[k] - mu) * rs * sg[idx] + sb[idx];
    hrow[k] = vv > 0.f ? vv : 0.f;
  }
  // Wave-synchronous LDS: DS ops from one wave stay in order, so the
  // cross-lane A-fragment reads below see the stores above.
  const float* A = sh[wave];
  v8f acc[4] = {};
  for (int kt = 0; kt < 32; ++kt) {
    const int kb = kt * 4 + half * 2;
    v2f a = *(const v2f*)(A + r * SA + kb);
    const float* w = sW + (kt * 2 + half) * SW + r * 2;
    v2f b[4];
#pragma unroll
    for (int nt = 0; nt < 4; ++nt) b[nt] = *(const v2f*)(w + nt * 32);
#pragma unroll
    for (int nt = 0; nt < 4; ++nt) acc[nt] = wmma4(a, b[nt], acc[nt]);
  }
#pragma unroll
  for (int nt = 0; nt < 4; ++nt)
#pragma unroll
    for (int v = 0; v < 8; ++v) {
      const int er = e0 + v + 8 * half;
      const int n = nt * 16 + r;
      eout[er * D + n] = ea[er * D + n] + acc[nt][v] + sbm[n];
    }
}

extern "C" void kernel_launch(void* const* d_in, const int* in_sizes, int n_in,
                              void* d_out, int out_size, void* d_ws, size_t ws_size,
                              hipStream_t stream) {
  const float* x   = (const float*)d_in[0];
  const int*   ei  = (const int*)d_in[1];
  const float* ea  = (const float*)d_in[2];
  const float* Wl  = (const float*)d_in[3];
  const float* Wr  = (const float*)d_in[4];
  const float* We  = (const float*)d_in[5];
  const float* att = (const float*)d_in[6];
  const float* cb  = (const float*)d_in[7];
  const float* lng = (const float*)d_in[8];
  const float* lnb = (const float*)d_in[9];
  const float* Wm  = (const float*)d_in[10];
  const float* bm  = (const float*)d_in[11];

  float* xout = (float*)d_out;              // [N, 64]
  float* eout = xout + (size_t)N * D;       // [E, 64]

  float* ws = (float*)d_ws;                 // ~40 MB used
  float* xl = ws;                           // N*64
  float* xr = xl + (size_t)N * D;           // N*64
  float* lg = xr + (size_t)N * D;           // E*4 (logits, then exp in place)
  int*   mx = (int*)(lg + (size_t)E * 4);   // N*4
  float* dn = (float*)(mx + (size_t)N * 4); // N*4

  k_init<<<(N * D + 255) / 256, 256, 0, stream>>>(xout, mx, dn, cb);
  k_node_gemm<<<(N / 16 + 7) / 8, 256, 0, stream>>>(x, Wl, Wr, xl, xr);
  k_edge_logits<<<E / 16 / 8, 256, 0, stream>>>(ea, ei, We, att, xl, xr, lg, mx);
  k_softmax_den<<<(E * 4 + 255) / 256, 256, 0, stream>>>(ei, lg, mx, dn);
  k_scatter<<<(E * D + 255) / 256, 256, 0, stream>>>(ei, lg, dn, xl, xout);
  k_edge_mlp<<<E / 16 / 4, 128, 0, stream>>>(ei, xout, ea, lng, lnb, Wm, bm, eout);
}